// KuramotoPIGNN_16535624090334
// MI455X (gfx1250) — compile-verified
//
#include <hip/hip_runtime.h>
#include <math.h>

typedef __attribute__((ext_vector_type(2))) float v2f;
typedef __attribute__((ext_vector_type(8))) float v8f;

#define HD 128
#define NLAYERS 3
#define BN_EPS 1e-5f

// ---------------- utility ----------------
__global__ void zero_f32(float* __restrict__ p, long n) {
    long i = (long)blockIdx.x * blockDim.x + threadIdx.x;
    if (i < n) p[i] = 0.0f;
}

// deg[d] += 1 per edge
__global__ void deg_accum(const long long* __restrict__ dst, float* __restrict__ deg, int E) {
    int e = blockIdx.x * blockDim.x + threadIdx.x;
    if (e < E) atomicAdd(&deg[(int)dst[e]], 1.0f);
}

__global__ void deg_to_dinv(float* __restrict__ deg, int N) {
    int i = blockIdx.x * blockDim.x + threadIdx.x;
    if (i < N) deg[i] = rsqrtf(deg[i] + 1.0f);
}

// ---------------- WMMA GEMM: Out[N,128] = A[N,128] @ W[128,128] (+bias)(+resid)(relu?) ----------------
// one wave32 computes one 16x16 output tile via V_WMMA_F32_16X16X4_F32, K-loop of 32 steps.
__global__ void gemm_wmma(const float* __restrict__ A, const float* __restrict__ W,
                          const float* __restrict__ bias, const float* __restrict__ resid,
                          float* __restrict__ Out, int Nrows, int do_relu) {
    const int lane = threadIdx.x & 31;
    const int wave = blockIdx.x * (blockDim.x >> 5) + (threadIdx.x >> 5);
    const int tiles_n = HD / 16;                 // 8 col tiles
    const int tm = wave / tiles_n;
    const int tn = wave % tiles_n;
    if (tm * 16 >= Nrows) return;                // uniform per-wave: EXEC stays all-ones

    const int mrow = tm * 16 + (lane & 15);      // A-frag row for this lane
    const int col  = tn * 16 + (lane & 15);      // B/C/D column for this lane
    const int kh   = (lane >> 4) << 1;           // lanes 0-15 hold K=k,k+1 ; lanes 16-31 hold K=k+2,k+3

    const float* __restrict__ arow = A + (long)mrow * HD;

    v8f acc = {};
    for (int k = 0; k < HD; k += 4) {
        v2f a, b;
        a.x = arow[k + kh];
        a.y = arow[k + kh + 1];
        b.x = W[(k + kh) * HD + col];
        b.y = W[(k + kh + 1) * HD + col];
        acc = __builtin_amdgcn_wmma_f32_16x16x4_f32(
            /*neg_a=*/false, a, /*neg_b=*/false, b,
            /*c_mod=*/(short)0, acc, /*reuse_a=*/false, /*reuse_b=*/false);
    }

    const float bcol = bias ? bias[col] : 0.0f;
#pragma unroll
    for (int v = 0; v < 8; ++v) {
        const int r = tm * 16 + v + ((lane >> 4) << 3);   // C/D layout: M = v + 8*(lane/16)
        float val = acc[v] + bcol;
        if (resid)   val += resid[(long)r * HD + col];
        if (do_relu) val = fmaxf(val, 0.0f);
        Out[(long)r * HD + col] = val;
    }
}

// ---------------- edge message pass: agg[dst] += hw[src] * dinv[src]*dinv[dst] ----------------
// one wave per edge; each lane moves a float4 (32 lanes * 4 = 128 features)
__global__ void edge_scatter(const float* __restrict__ hw,
                             const long long* __restrict__ src,
                             const long long* __restrict__ dst,
                             const float* __restrict__ dinv,
                             float* __restrict__ agg, int E) {
    const int lane = threadIdx.x & 31;
    const int e = blockIdx.x * (blockDim.x >> 5) + (threadIdx.x >> 5);
    if (e >= E) return;
    const int s = (int)src[e];
    const int d = (int)dst[e];
    const float norm = dinv[s] * dinv[d];
    const float4 v = ((const float4*)(hw + (long)s * HD))[lane];
    float* ap = agg + (long)d * HD + lane * 4;
    atomicAdd(ap + 0, v.x * norm);
    atomicAdd(ap + 1, v.y * norm);
    atomicAdd(ap + 2, v.z * norm);
    atomicAdd(ap + 3, v.w * norm);
}

// ---------------- self-loop + conv bias + batchnorm + relu ----------------
__global__ void post_bn_relu(const float* __restrict__ agg, const float* __restrict__ hw,
                             const float* __restrict__ dinv, const float* __restrict__ cb,
                             const float* __restrict__ gamma, const float* __restrict__ beta,
                             const float* __restrict__ mean, const float* __restrict__ var,
                             float* __restrict__ out, int N) {
    long i = (long)blockIdx.x * blockDim.x + threadIdx.x;
    if (i >= (long)N * HD) return;
    const int n = (int)(i / HD);
    const int f = (int)(i % HD);
    const float di = dinv[n];
    float val = agg[i] + hw[i] * di * di + cb[f];
    val = (val - mean[f]) * rsqrtf(var[f] + BN_EPS) * gamma[f] + beta[f];
    out[i] = fmaxf(val, 0.0f);
}

// ---------------- fold global-context rows of fus_W1 into an effective bias ----------------
__global__ void fusion_bias(const float* __restrict__ W1, const float* __restrict__ b1,
                            const float* __restrict__ g, float* __restrict__ fbias) {
    const int f = threadIdx.x;   // 128 threads
    float v = b1[f];
#pragma unroll
    for (int j = 0; j < 4; ++j) v += g[j] * W1[(HD + j) * HD + f];
    fbias[f] = v;
}

// ---------------- decoder tail: delta = z @ W2 + b2 ; theta_next = atan2(sin,cos) ----------------
// one wave per node, lane-strided dot + butterfly reduction
__global__ void decode(const float* __restrict__ z, const float* __restrict__ W2,
                       const float* __restrict__ b2, const float* __restrict__ theta,
                       float* __restrict__ out, int N) {
    const int lane = threadIdx.x & 31;
    const int n = blockIdx.x * (blockDim.x >> 5) + (threadIdx.x >> 5);
    if (n >= N) return;
    const float* zr = z + (long)n * HD;
    float sum = 0.0f;
#pragma unroll
    for (int i = lane; i < HD; i += 32) sum += zr[i] * W2[i];
#pragma unroll
    for (int off = 16; off; off >>= 1) sum += __shfl_xor(sum, off, 32);
    if (lane == 0) {
        const float delta = sum + b2[0];
        out[n] = delta;
        const float s = theta[n] + delta;
        out[N + n] = atan2f(sinf(s), cosf(s));
    }
}

// ---------------- host driver ----------------
extern "C" void kernel_launch(void* const* d_in, const int* in_sizes, int n_in,
                              void* d_out, int out_size, void* d_ws, size_t ws_size,
                              hipStream_t stream) {
    const float*     x     = (const float*)d_in[0];
    const long long* ei    = (const long long*)d_in[1];
    const float*     gctx  = (const float*)d_in[2];
    const float*     theta = (const float*)d_in[3];
    const float*     convW = (const float*)d_in[4];
    const float*     convB = (const float*)d_in[5];
    const float*     gma   = (const float*)d_in[6];
    const float*     bta   = (const float*)d_in[7];
    const float*     mu    = (const float*)d_in[8];
    const float*     var   = (const float*)d_in[9];
    const float*     fW1   = (const float*)d_in[10];
    const float*     fb1   = (const float*)d_in[11];
    const float*     fW2   = (const float*)d_in[12];
    const float*     fb2   = (const float*)d_in[13];
    const float*     dW1   = (const float*)d_in[14];
    const float*     db1   = (const float*)d_in[15];
    const float*     dW2   = (const float*)d_in[16];
    const float*     db2   = (const float*)d_in[17];
    float*           out   = (float*)d_out;

    const int N = in_sizes[0] / HD;
    const int E = in_sizes[1] / 2;
    const long long* src = ei;
    const long long* dst = ei + E;

    // workspace partition: 3 NxH matrices + dinv + effective fusion bias
    char* ws = (char*)d_ws;
    const size_t matBytes = (size_t)N * HD * sizeof(float);
    float* bufA  = (float*)(ws);                  // h (post-layer), residual
    float* bufB  = (float*)(ws + matBytes);       // hw / z
    float* bufC  = (float*)(ws + 2 * matBytes);   // agg / fused h
    float* dinv  = (float*)(ws + 3 * matBytes);   // N floats (deg then dinv)
    float* fbias = dinv + N;                      // 128 floats

    const long NH = (long)N * HD;
    const int elemBlocks = (int)((NH + 255) / 256);

    // degrees -> dinv
    zero_f32<<<(N + 255) / 256, 256, 0, stream>>>(dinv, N);
    deg_accum<<<(E + 255) / 256, 256, 0, stream>>>(dst, dinv, E);
    deg_to_dinv<<<(N + 255) / 256, 256, 0, stream>>>(dinv, N);

    const int tilesM = (N + 15) / 16;
    const int gemmBlocks = (tilesM * (HD / 16) + 3) / 4;   // 4 waves (128 thr) per block
    const int edgeBlocks = (E + 3) / 4;                    // 1 wave per edge
    const int decBlocks  = (N + 3) / 4;                    // 1 wave per node

    // ---- 3 GCN layers ----
    const float* hcur = x;
    for (int l = 0; l < NLAYERS; ++l) {
        gemm_wmma<<<gemmBlocks, 128, 0, stream>>>(hcur, convW + (size_t)l * HD * HD,
                                                  nullptr, nullptr, bufB, N, 0);
        zero_f32<<<elemBlocks, 256, 0, stream>>>(bufC, NH);
        edge_scatter<<<edgeBlocks, 128, 0, stream>>>(bufB, src, dst, dinv, bufC, E);
        post_bn_relu<<<elemBlocks, 256, 0, stream>>>(bufC, bufB, dinv,
                                                     convB + l * HD, gma + l * HD, bta + l * HD,
                                                     mu + l * HD, var + l * HD, bufA, N);
        hcur = bufA;
    }

    // ---- global-context fusion: h = h + relu(cat(h,g) @ W1 + b1) @ W2 + b2 ----
    fusion_bias<<<1, HD, 0, stream>>>(fW1, fb1, gctx, fbias);
    gemm_wmma<<<gemmBlocks, 128, 0, stream>>>(bufA, fW1, fbias, nullptr, bufB, N, 1); // z1
    gemm_wmma<<<gemmBlocks, 128, 0, stream>>>(bufB, fW2, fb2, bufA, bufC, N, 0);      // h + z1@W2 + b2

    // ---- decoder ----
    gemm_wmma<<<gemmBlocks, 128, 0, stream>>>(bufC, dW1, db1, nullptr, bufB, N, 1);   // relu(h@dW1+db1)
    decode<<<decBlocks, 128, 0, stream>>>(bufB, dW2, db2, theta, out, N);
}